// FSABlock_19542101197001
// MI455X (gfx1250) — compile-verified
//
#include <hip/hip_runtime.h>
#include <math.h>

#define C_DIM 384
#define HW_DIM 4096
#define IMG_W 64

typedef __attribute__((ext_vector_type(2))) float v2f;
typedef __attribute__((ext_vector_type(8))) float v8f;

// D = A(16x4 f32) * B(4x16 f32) + C(16x16 f32), one matrix per wave32.
__device__ __forceinline__ v8f wmma_f32(v2f a, v2f b, v8f c) {
    return __builtin_amdgcn_wmma_f32_16x16x4_f32(
        /*neg_a=*/false, a, /*neg_b=*/false, b,
        /*c_mod=*/(short)0, c, /*reuse_a=*/false, /*reuse_b=*/false);
}

// Safe zero-padded load: always reads an in-bounds address, selects 0 if !ok.
__device__ __forceinline__ float ldz(const float* p, int off, bool ok) {
    float v = p[ok ? off : 0];
    return ok ? v : 0.0f;
}

// ---------------------------------------------------------------------------
// GroupNorm: one block per (batch, group). C=384, 32 groups -> 12 ch/group.
// ---------------------------------------------------------------------------
__global__ __launch_bounds__(256)
void groupnorm_kernel(const float* __restrict__ x, const float* __restrict__ gamma,
                      const float* __restrict__ beta, float* __restrict__ xn) {
    __shared__ float s0[256];
    __shared__ float s1[256];
    const int tid = threadIdx.x;
    const int b = blockIdx.x >> 5;
    const int g = blockIdx.x & 31;
    const size_t base = ((size_t)b * C_DIM + g * 12) * HW_DIM;
    const int N = 12 * HW_DIM;  // 49152
    float sum = 0.0f, sq = 0.0f;
    for (int i = tid; i < N; i += 256) {
        float v = x[base + i];
        sum += v;
        sq += v * v;
    }
    s0[tid] = sum; s1[tid] = sq;
    __syncthreads();
    for (int off = 128; off > 0; off >>= 1) {
        if (tid < off) { s0[tid] += s0[tid + off]; s1[tid] += s1[tid + off]; }
        __syncthreads();
    }
    const float mu = s0[0] * (1.0f / (float)N);
    const float var = s1[0] * (1.0f / (float)N) - mu * mu;
    const float inv = rsqrtf(var + 1e-5f);
    for (int i = tid; i < N; i += 256) {
        const int c = g * 12 + (i >> 12);
        xn[base + i] = (x[base + i] - mu) * inv * gamma[c] + beta[c];
    }
}

// ---------------------------------------------------------------------------
// 1x1 conv = GEMM: Y[b][m][n] = sum_k W[m][k] * X[b][k][n] + bias[m]
// Optionally multiplied elementwise by `mul` (for the final xn*res fusion).
// Wave tile: M=16, N=64 (4 WMMA accumulators), K-step 4.
// grid = (HW/512, C/16, B), block = (32, 8)
// ---------------------------------------------------------------------------
__global__ __launch_bounds__(256)
void conv1x1_wmma_kernel(const float* __restrict__ X, const float* __restrict__ W,
                         const float* __restrict__ bias, float* __restrict__ Y,
                         const float* __restrict__ mul) {
    const int lane = threadIdx.x;
    const int lm = lane & 15;
    const int hi = lane >> 4;
    const int b = blockIdx.z;
    const int m0 = blockIdx.y * 16;
    const int n0 = blockIdx.x * 512 + threadIdx.y * 64;
    const float* Xb = X + (size_t)b * C_DIM * HW_DIM;
    v8f acc0 = {}, acc1 = {}, acc2 = {}, acc3 = {};
    for (int k0 = 0; k0 < C_DIM; k0 += 4) {
        const int ka = k0 + 2 * hi;  // A 16x4 layout: lanes 0-15 K={0,1}, 16-31 K={2,3}
        v2f a;
        a.x = W[(size_t)(m0 + lm) * C_DIM + ka];
        a.y = W[(size_t)(m0 + lm) * C_DIM + ka + 1];
        const float* r0 = Xb + (size_t)ka * HW_DIM;
        const float* r1 = r0 + HW_DIM;
        v2f b0, b1, b2, b3;
        b0.x = r0[n0 + lm];      b0.y = r1[n0 + lm];
        b1.x = r0[n0 + 16 + lm]; b1.y = r1[n0 + 16 + lm];
        b2.x = r0[n0 + 32 + lm]; b2.y = r1[n0 + 32 + lm];
        b3.x = r0[n0 + 48 + lm]; b3.y = r1[n0 + 48 + lm];
        acc0 = wmma_f32(a, b0, acc0);
        acc1 = wmma_f32(a, b1, acc1);
        acc2 = wmma_f32(a, b2, acc2);
        acc3 = wmma_f32(a, b3, acc3);
    }
    float* Yb = Y + (size_t)b * C_DIM * HW_DIM;
    const float* Mb = mul ? (mul + (size_t)b * C_DIM * HW_DIM) : nullptr;
#pragma unroll
    for (int r = 0; r < 8; ++r) {
        const int m = m0 + r + 8 * hi;  // C/D layout: VGPR r -> M=r (+8 for hi lanes)
        const float bv = bias[m];
        const size_t base = (size_t)m * HW_DIM + n0 + lm;
        float o0 = acc0[r] + bv;
        float o1 = acc1[r] + bv;
        float o2 = acc2[r] + bv;
        float o3 = acc3[r] + bv;
        if (Mb) {
            o0 *= Mb[base];
            o1 *= Mb[base + 16];
            o2 *= Mb[base + 32];
            o3 *= Mb[base + 48];
        }
        Yb[base]      = o0;
        Yb[base + 16] = o1;
        Yb[base + 32] = o2;
        Yb[base + 48] = o3;
    }
}

// ---------------------------------------------------------------------------
// 3x3 conv (pad=1) as implicit GEMM. N-tile = 64 consecutive pixels of one
// image row (n0 is a multiple of 64 -> constant y per wave). Out-of-range
// rows skip the whole tap; x handled per-lane with zero-fill selects.
// ---------------------------------------------------------------------------
__global__ __launch_bounds__(256)
void conv3x3_wmma_kernel(const float* __restrict__ X, const float* __restrict__ W,
                         const float* __restrict__ bias, float* __restrict__ Y) {
    const int lane = threadIdx.x;
    const int lm = lane & 15;
    const int hi = lane >> 4;
    const int b = blockIdx.z;
    const int m0 = blockIdx.y * 16;
    const int n0 = blockIdx.x * 512 + threadIdx.y * 64;
    const int y = n0 >> 6;
    const float* Xb = X + (size_t)b * C_DIM * HW_DIM;
    v8f acc0 = {}, acc1 = {}, acc2 = {}, acc3 = {};
    for (int tap = 0; tap < 9; ++tap) {
        const int dy = tap / 3 - 1;
        const int dx = tap % 3 - 1;
        const int ys = y + dy;
        if ((unsigned)ys >= (unsigned)IMG_W) continue;  // zero-padded row
        const int x0 = lm + dx;
        const int x1 = 16 + lm + dx;
        const int x2 = 32 + lm + dx;
        const int x3 = 48 + lm + dx;
        const bool ok0 = (unsigned)x0 < (unsigned)IMG_W;
        const bool ok3 = (unsigned)x3 < (unsigned)IMG_W;
        for (int k0 = 0; k0 < C_DIM; k0 += 4) {
            const int ka = k0 + 2 * hi;
            v2f a;
            a.x = W[((size_t)(m0 + lm) * C_DIM + ka) * 9 + tap];
            a.y = W[((size_t)(m0 + lm) * C_DIM + ka + 1) * 9 + tap];
            const float* r0 = Xb + (size_t)ka * HW_DIM + ys * IMG_W;
            const float* r1 = r0 + HW_DIM;
            v2f b0, b1, b2, b3;
            b0.x = ldz(r0, x0, ok0); b0.y = ldz(r1, x0, ok0);
            b1.x = r0[x1];           b1.y = r1[x1];   // 1..62 always in range? x1 in [15..17+?]: 16+lm+dx in [15,62] -> in range
            b2.x = r0[x2];           b2.y = r1[x2];   // [31,... ,49] in range
            b3.x = ldz(r0, x3, ok3); b3.y = ldz(r1, x3, ok3);
            acc0 = wmma_f32(a, b0, acc0);
            acc1 = wmma_f32(a, b1, acc1);
            acc2 = wmma_f32(a, b2, acc2);
            acc3 = wmma_f32(a, b3, acc3);
        }
    }
    float* Yb = Y + (size_t)b * C_DIM * HW_DIM;
#pragma unroll
    for (int r = 0; r < 8; ++r) {
        const int m = m0 + r + 8 * hi;
        const float bv = bias[m];
        const size_t base = (size_t)m * HW_DIM + n0 + lm;
        Yb[base]      = acc0[r] + bv;
        Yb[base + 16] = acc1[r] + bv;
        Yb[base + 32] = acc2[r] + bv;
        Yb[base + 48] = acc3[r] + bv;
    }
}

// ---------------------------------------------------------------------------
// 64-point radix-2 DIF FFT passes over rows/columns of a 64x64 complex LDS
// image. sign=-1: forward DFT; sign=+1: unnormalized inverse.
// ---------------------------------------------------------------------------
__device__ void fft_lines(float* re, float* im, int tid, float sign, bool rows) {
    for (int s = 5; s >= 0; --s) {
        const int h = 1 << s;
        for (int item = tid; item < 2048; item += 256) {
            const int line = item >> 5;
            const int j = item & 31;
            const int p = j & (h - 1);
            const int grp = j >> s;
            const int i0 = (grp << (s + 1)) + p;
            const int i1 = i0 + h;
            const int idx0 = rows ? (line * 64 + i0) : (i0 * 64 + line);
            const int idx1 = rows ? (line * 64 + i1) : (i1 * 64 + line);
            const float ar = re[idx0], ai = im[idx0];
            const float br = re[idx1], bi = im[idx1];
            const float ang = sign * 3.14159265358979323846f * (float)p / (float)h;
            float sn, cs;
            __sincosf(ang, &sn, &cs);
            re[idx0] = ar + br;
            im[idx0] = ai + bi;
            const float tr = ar - br, ti = ai - bi;
            re[idx1] = tr * cs - ti * sn;
            im[idx1] = tr * sn + ti * cs;
        }
        __syncthreads();
    }
    // bit-reverse permutation along the transformed dimension
    for (int item = tid; item < 4096; item += 256) {
        const int line = item >> 6;
        const int e = item & 63;
        const int r = ((e & 1) << 5) | ((e & 2) << 3) | ((e & 4) << 1) |
                      ((e & 8) >> 1) | ((e & 16) >> 3) | ((e & 32) >> 5);
        if (r > e) {
            const int iA = rows ? (line * 64 + e) : (e * 64 + line);
            const int iB = rows ? (line * 64 + r) : (r * 64 + line);
            float t;
            t = re[iA]; re[iA] = re[iB]; re[iB] = t;
            t = im[iA]; im[iA] = im[iB]; im[iB] = t;
        }
    }
    __syncthreads();
}

__device__ void fft2_inplace(float* re, float* im, int tid, float sign) {
    fft_lines(re, im, tid, sign, true);
    fft_lines(re, im, tid, sign, false);
}

// ---------------------------------------------------------------------------
// A = ifft2(fft2(q) * fft2(k)).real per (B,C) image, via the packed trick
// Z = fft2(q + i*k);  Fq = (Z+conj(Z-))/2,  Fk = (Z-conj(Z-))/(2i).
// One block per image. May write A over the q buffer (fully staged in LDS).
// ---------------------------------------------------------------------------
__global__ __launch_bounds__(256)
void fft_corr_kernel(const float* Q, const float* K, float* A) {
    __shared__ float sre[4096];
    __shared__ float sim[4096];
    const int tid = threadIdx.x;
    const size_t img = (size_t)blockIdx.x * HW_DIM;
    for (int i = tid; i < 4096; i += 256) {
        sre[i] = Q[img + i];
        sim[i] = K[img + i];
    }
    __syncthreads();
    fft2_inplace(sre, sim, tid, -1.0f);  // forward
    // Pointwise product in frequency domain via conjugate-symmetric pairing.
    for (int i = tid; i < 4096; i += 256) {
        const int u = i >> 6, v = i & 63;
        const int um = (64 - u) & 63, vm = (64 - v) & 63;
        const int mi = (um << 6) | vm;
        if (mi < i) continue;  // each pair handled once
        const float zr = sre[i],  zi = sim[i];
        const float wr = sre[mi], wi = sim[mi];
        // At index i:  Fq = 0.5(z + conj(w)),  Fk = (z - conj(w)) / (2i)
        const float fqr = 0.5f * (zr + wr), fqi = 0.5f * (zi - wi);
        const float fkr = 0.5f * (zi + wi), fki = 0.5f * (wr - zr);
        const float pr = fqr * fkr - fqi * fki;
        const float pi = fqr * fki + fqi * fkr;
        // At index mi: Fq = 0.5(w + conj(z)),  Fk = (w - conj(z)) / (2i)
        const float gqr = 0.5f * (wr + zr), gqi = 0.5f * (wi - zi);
        const float gkr = 0.5f * (wi + zi), gki = 0.5f * (zr - wr);
        const float qr2 = gqr * gkr - gqi * gki;
        const float qi2 = gqr * gki + gqi * gkr;
        sre[i] = pr; sim[i] = pi;
        if (mi != i) { sre[mi] = qr2; sim[mi] = qi2; }
    }
    __syncthreads();
    fft2_inplace(sre, sim, tid, +1.0f);  // unnormalized inverse
    for (int i = tid; i < 4096; i += 256)
        A[img + i] = sre[i] * (1.0f / 4096.0f);
}

// ---------------------------------------------------------------------------
// Per-(B,C) row: exact k-th-largest threshold by 32-bit radix select on
// order-preserving float keys, then masked softmax fused with F4 = v * Fw.
// One block per row; each thread caches 16 values in registers.
// ---------------------------------------------------------------------------
__global__ __launch_bounds__(256)
void topk_softmax_kernel(const float* __restrict__ A, const float* __restrict__ V,
                         const int* __restrict__ kptr, float* __restrict__ F4) {
    __shared__ int si[256];
    __shared__ float sf[256];
    const int tid = threadIdx.x;
    const size_t row = (size_t)blockIdx.x * HW_DIM;
    const int kk = kptr[0];
    float xv[16];
    unsigned key[16];
#pragma unroll
    for (int i = 0; i < 16; ++i) {
        const float f = A[row + tid + i * 256];
        xv[i] = f;
        const unsigned u = __float_as_uint(f);
        key[i] = (u & 0x80000000u) ? ~u : (u | 0x80000000u);  // monotone map
    }
    unsigned prefix = 0u;
    for (int bit = 31; bit >= 0; --bit) {
        const unsigned cand = prefix | (1u << bit);
        int c = 0;
#pragma unroll
        for (int i = 0; i < 16; ++i) c += (key[i] >= cand) ? 1 : 0;
        si[tid] = c;
        __syncthreads();
        for (int off = 128; off > 0; off >>= 1) {
            if (tid < off) si[tid] += si[tid + off];
            __syncthreads();
        }
        const int total = si[0];
        __syncthreads();
        if (total >= kk) prefix = cand;  // same decision on every thread
    }
    const unsigned tu = (prefix & 0x80000000u) ? (prefix ^ 0x80000000u) : ~prefix;
    const float thresh = __uint_as_float(tu);
    // row max (== max over selected)
    float m = -INFINITY;
#pragma unroll
    for (int i = 0; i < 16; ++i) m = fmaxf(m, xv[i]);
    sf[tid] = m;
    __syncthreads();
    for (int off = 128; off > 0; off >>= 1) {
        if (tid < off) sf[tid] = fmaxf(sf[tid], sf[tid + off]);
        __syncthreads();
    }
    m = sf[0];
    __syncthreads();
    // sum of exp over selected
    float s = 0.0f;
#pragma unroll
    for (int i = 0; i < 16; ++i)
        if (xv[i] >= thresh) s += __expf(xv[i] - m);
    sf[tid] = s;
    __syncthreads();
    for (int off = 128; off > 0; off >>= 1) {
        if (tid < off) sf[tid] += sf[tid + off];
        __syncthreads();
    }
    const float inv = 1.0f / sf[0];
#pragma unroll
    for (int i = 0; i < 16; ++i) {
        const float fw = (xv[i] >= thresh) ? __expf(xv[i] - m) * inv : 0.0f;
        F4[row + tid + i * 256] = V[row + tid + i * 256] * fw;
    }
}

// ---------------------------------------------------------------------------
// Orchestration. Workspace layout (floats): [xn | q | k | v], 4 x 50.33 MB.
// Reuse: A overwrites q (FFT stages whole image in LDS first); F4 overwrites k.
// ---------------------------------------------------------------------------
extern "C" void kernel_launch(void* const* d_in, const int* in_sizes, int n_in,
                              void* d_out, int out_size, void* d_ws, size_t ws_size,
                              hipStream_t stream) {
    (void)in_sizes; (void)n_in; (void)out_size; (void)ws_size;
    const float* x     = (const float*)d_in[0];
    const float* gamma = (const float*)d_in[1];
    const float* beta  = (const float*)d_in[2];
    const float* wq    = (const float*)d_in[3];
    const float* bq    = (const float*)d_in[4];
    const float* wk    = (const float*)d_in[5];
    const float* bk    = (const float*)d_in[6];
    const float* wv    = (const float*)d_in[7];
    const float* bv    = (const float*)d_in[8];
    const float* wf    = (const float*)d_in[9];
    const float* bf    = (const float*)d_in[10];
    const int*   kptr  = (const int*)d_in[11];
    float* out = (float*)d_out;

    const size_t IMG = (size_t)8 * C_DIM * HW_DIM;
    float* xn = (float*)d_ws;
    float* qb = xn + IMG;
    float* kb = qb + IMG;
    float* vb = kb + IMG;

    const dim3 gemm_grid(HW_DIM / 512, C_DIM / 16, 8);
    const dim3 gemm_block(32, 8);

    groupnorm_kernel<<<dim3(8 * 32), dim3(256), 0, stream>>>(x, gamma, beta, xn);
    conv1x1_wmma_kernel<<<gemm_grid, gemm_block, 0, stream>>>(xn, wq, bq, qb, nullptr);
    conv1x1_wmma_kernel<<<gemm_grid, gemm_block, 0, stream>>>(xn, wv, bv, vb, nullptr);
    conv3x3_wmma_kernel<<<gemm_grid, gemm_block, 0, stream>>>(xn, wk, bk, kb);
    fft_corr_kernel<<<dim3(8 * C_DIM), dim3(256), 0, stream>>>(qb, kb, qb);      // A -> qb
    topk_softmax_kernel<<<dim3(8 * C_DIM), dim3(256), 0, stream>>>(qb, vb, kptr, kb);  // F4 -> kb
    conv1x1_wmma_kernel<<<gemm_grid, gemm_block, 0, stream>>>(kb, wf, bf, out, xn);    // out = xn * (conv+bias)
}